// CellTypeGNN_28363964023038
// MI455X (gfx1250) — compile-verified
//
#include <hip/hip_runtime.h>
#include <hip/hip_bf16.h>

typedef __attribute__((ext_vector_type(2))) float v2f;
typedef __attribute__((ext_vector_type(8))) float v8f;

#define D_H 128

__device__ __forceinline__ float wave_sum32(float v) {
    #pragma unroll
    for (int m = 16; m >= 1; m >>= 1) v += __shfl_xor(v, m, 32);
    return v;
}

__device__ __forceinline__ float gelu_exact(float x) {
    return 0.5f * x * (1.0f + erff(x * 0.7071067811865475f));
}

// ---------------- degree / aggregation ----------------

__global__ __launch_bounds__(256) void deg_count_k(const int* __restrict__ dst,
                                                   float* __restrict__ deg, int E) {
    int i = blockIdx.x * 256 + threadIdx.x;
    if (i < E) atomicAdd(&deg[dst[i]], 1.0f);
}

__global__ __launch_bounds__(256) void deg_inv_k(float* __restrict__ deg, int N) {
    int i = blockIdx.x * 256 + threadIdx.x;
    if (i < N) deg[i] = 1.0f / fmaxf(deg[i], 1.0f);
}

// one edge handled by 32 threads, 4 floats each (128 features)
__global__ __launch_bounds__(256) void agg_edges_k(const float* __restrict__ x,
                                                   const int* __restrict__ src,
                                                   const int* __restrict__ dst,
                                                   float* __restrict__ agg, int E) {
    long long gid = (long long)blockIdx.x * 256 + threadIdx.x;
    int e = (int)(gid >> 5);
    if (e >= E) return;
    int sub = ((int)gid & 31) * 4;
    int s = src[e], d = dst[e];
    const float4 v = *reinterpret_cast<const float4*>(x + (size_t)s * D_H + sub);
    float* a = agg + (size_t)d * D_H + sub;
    atomicAdd(a + 0, v.x);
    atomicAdd(a + 1, v.y);
    atomicAdd(a + 2, v.z);
    atomicAdd(a + 3, v.w);
}

// ---------------- weight transpose: WT[c*128 + k] = W[k*dout + c] ----------

__global__ __launch_bounds__(256) void transpose_w_k(const float* __restrict__ W,
                                                     float* __restrict__ WT,
                                                     int dout) {
    int i = blockIdx.x * 256 + threadIdx.x;   // over 128*dout elements
    if (i < D_H * dout) {
        int k = i / dout, c = i % dout;
        WT[(size_t)c * D_H + k] = W[i];
    }
}

// ---------------- fused SAGE GEMM via V_WMMA_F32_16X16X4_F32 ----------------
// out[n, :dout] = (agg[n,:]*invdeg[n]) @ Wl + bl + xin[n,:] @ Wr
// Weights are pre-transposed (column-major, ld=128) so every fragment load is
// a contiguous float2 with an immediate offset from a loop-invariant base.
// 4 waves/block; wave w does one 16x16 tile at col blockIdx.y*64 + w*16.

__global__ __launch_bounds__(128) void sage_gemm_k(const float* __restrict__ agg,
                                                   const float* __restrict__ invdeg,
                                                   const float* __restrict__ xin,
                                                   const float* __restrict__ WlT,
                                                   const float* __restrict__ bl,
                                                   const float* __restrict__ WrT,
                                                   float* __restrict__ out,
                                                   int nrows, int dout) {
    const int lane = threadIdx.x & 31;
    const int wave = threadIdx.x >> 5;
    const int rowBase = blockIdx.x * 16;
    const int m = lane & 15;                       // A row within tile / D col lane
    const int khalf = (lane >> 4) * 2;             // lanes 16-31 carry K+2,K+3
    const int col = blockIdx.y * 64 + wave * 16 + m;

    int row = rowBase + m;
    if (row > nrows - 1) row = nrows - 1;          // clamp: keep EXEC all-ones for WMMA
    const float idg = invdeg[row];

    // four loop-invariant, 8B-aligned base pointers; inner loop uses imm offsets
    const float* aggP = agg + (size_t)row * D_H + khalf;
    const float* xP   = xin + (size_t)row * D_H + khalf;
    const float* wlP  = WlT + (size_t)col * D_H + khalf;
    const float* wrP  = WrT + (size_t)col * D_H + khalf;

    v8f c = {};
    #pragma unroll
    for (int k0 = 0; k0 < D_H; k0 += 4) {
        v2f aA = *reinterpret_cast<const v2f*>(aggP + k0) * idg;
        v2f aX = *reinterpret_cast<const v2f*>(xP + k0);
        v2f bL = *reinterpret_cast<const v2f*>(wlP + k0);
        v2f bR = *reinterpret_cast<const v2f*>(wrP + k0);
        c = __builtin_amdgcn_wmma_f32_16x16x4_f32(false, aA, false, bL, (short)0, c, false, false);
        c = __builtin_amdgcn_wmma_f32_16x16x4_f32(false, aX, false, bR, (short)0, c, false, false);
    }

    const float bias = bl[col];
    const int rOut = rowBase + (lane >> 4) * 8;    // D: VGPR r holds rows r / r+8
    #pragma unroll
    for (int r = 0; r < 8; ++r) {
        int orow = rOut + r;
        if (orow < nrows) out[(size_t)orow * dout + col] = c[r] + bias;
    }
}

// ---------------- LayerNorm + exact GELU + residual (D=128, wave per row) ---

__global__ __launch_bounds__(256) void ln_gelu_res_k(const float* __restrict__ h,
                                                     const float* __restrict__ res,
                                                     const float* __restrict__ g,
                                                     const float* __restrict__ b,
                                                     float* __restrict__ out, int nrows) {
    const int lane = threadIdx.x & 31;
    const int row = blockIdx.x * 8 + (threadIdx.x >> 5);
    if (row >= nrows) return;
    const float* hr = h + (size_t)row * D_H;
    float v[4];
    float s = 0.0f;
    #pragma unroll
    for (int j = 0; j < 4; ++j) { v[j] = hr[lane + 32 * j]; s += v[j]; }
    s = wave_sum32(s);
    const float mu = s * (1.0f / 128.0f);
    float q = 0.0f;
    #pragma unroll
    for (int j = 0; j < 4; ++j) { float d = v[j] - mu; q += d * d; }
    q = wave_sum32(q);
    const float rs = rsqrtf(q * (1.0f / 128.0f) + 1e-5f);
    #pragma unroll
    for (int j = 0; j < 4; ++j) {
        const int idx = lane + 32 * j;
        float y = (v[j] - mu) * rs * g[idx] + b[idx];
        out[(size_t)row * D_H + idx] = gelu_exact(y) + res[(size_t)row * D_H + idx];
    }
}

__global__ __launch_bounds__(256) void gelu_ip_k(float* __restrict__ y, long long n) {
    long long i = (long long)blockIdx.x * 256 + threadIdx.x;
    if (i < n) y[i] = gelu_exact(y[i]);
}

// ---------------- classifier: LN(64) then @ Wc[64,40] + bcls ----------------

__global__ __launch_bounds__(256) void classifier_k(const float* __restrict__ y,
                                                    const float* __restrict__ gc,
                                                    const float* __restrict__ bc,
                                                    const float* __restrict__ Wc,
                                                    const float* __restrict__ bcls,
                                                    float* __restrict__ out, int nrows) {
    __shared__ float sdata[8][64];
    const int lane = threadIdx.x & 31;
    const int wv = threadIdx.x >> 5;
    const int row = blockIdx.x * 8 + wv;
    float v0 = 0.0f, v1 = 0.0f;
    if (row < nrows) {
        v0 = y[(size_t)row * 64 + lane];
        v1 = y[(size_t)row * 64 + 32 + lane];
    }
    const float mu = wave_sum32(v0 + v1) * (1.0f / 64.0f);
    const float d0 = v0 - mu, d1 = v1 - mu;
    const float rs = rsqrtf(wave_sum32(d0 * d0 + d1 * d1) * (1.0f / 64.0f) + 1e-5f);
    sdata[wv][lane]      = d0 * rs * gc[lane]      + bc[lane];
    sdata[wv][32 + lane] = d1 * rs * gc[32 + lane] + bc[32 + lane];
    __syncthreads();
    if (row < nrows) {
        for (int cc = lane; cc < 40; cc += 32) {
            float acc = bcls[cc];
            #pragma unroll
            for (int k = 0; k < 64; ++k) acc += sdata[wv][k] * Wc[k * 40 + cc];
            out[(size_t)row * 40 + cc] = acc;
        }
    }
}

// ---------------------------------------------------------------------------

extern "C" void kernel_launch(void* const* d_in, const int* in_sizes, int n_in,
                              void* d_out, int out_size, void* d_ws, size_t ws_size,
                              hipStream_t stream) {
    const float* x   = (const float*)d_in[0];
    const int*   ei  = (const int*)d_in[1];
    const float* Wl1 = (const float*)d_in[2];
    const float* bl1 = (const float*)d_in[3];
    const float* Wr1 = (const float*)d_in[4];
    const float* g1  = (const float*)d_in[5];
    const float* b1  = (const float*)d_in[6];
    const float* Wl2 = (const float*)d_in[7];
    const float* bl2 = (const float*)d_in[8];
    const float* Wr2 = (const float*)d_in[9];
    const float* g2  = (const float*)d_in[10];
    const float* b2  = (const float*)d_in[11];
    const float* Wl3 = (const float*)d_in[12];
    const float* bl3 = (const float*)d_in[13];
    const float* Wr3 = (const float*)d_in[14];
    const float* gc  = (const float*)d_in[15];
    const float* bc  = (const float*)d_in[16];
    const float* Wc  = (const float*)d_in[17];
    const float* bcls = (const float*)d_in[18];

    const int E = in_sizes[1] / 2;
    const int N = in_sizes[0] / D_H;
    const int* src = ei;
    const int* dst = ei + E;

    float* ws   = (float*)d_ws;
    float* aggB = ws;                                  // N*128
    float* hB   = aggB + (size_t)N * D_H;              // N*128
    float* x1   = hB + (size_t)N * D_H;                // N*128 (also y with ld=64)
    float* deg  = x1 + (size_t)N * D_H;                // N
    float* WlT1 = deg + N;                             // 128*128
    float* WrT1 = WlT1 + 128 * 128;
    float* WlT2 = WrT1 + 128 * 128;
    float* WrT2 = WlT2 + 128 * 128;
    float* WlT3 = WrT2 + 128 * 128;                    // 128*64
    float* WrT3 = WlT3 + 128 * 64;

    const size_t featBytes = (size_t)N * D_H * sizeof(float);
    const int rowBlocks = (N + 15) / 16;
    const int rowWaves8 = (N + 7) / 8;
    const int edgeLaneBlocks = (int)(((long long)E * 32 + 255) / 256);

    // pre-transpose weights (column-major, ld=128) for contiguous B fragments
    transpose_w_k<<<(128 * 128 + 255) / 256, 256, 0, stream>>>(Wl1, WlT1, 128);
    transpose_w_k<<<(128 * 128 + 255) / 256, 256, 0, stream>>>(Wr1, WrT1, 128);
    transpose_w_k<<<(128 * 128 + 255) / 256, 256, 0, stream>>>(Wl2, WlT2, 128);
    transpose_w_k<<<(128 * 128 + 255) / 256, 256, 0, stream>>>(Wr2, WrT2, 128);
    transpose_w_k<<<(128 * 64 + 255) / 256, 256, 0, stream>>>(Wl3, WlT3, 64);
    transpose_w_k<<<(128 * 64 + 255) / 256, 256, 0, stream>>>(Wr3, WrT3, 64);

    // degrees (shared by all three aggregations)
    hipMemsetAsync(deg, 0, (size_t)N * sizeof(float), stream);
    deg_count_k<<<(E + 255) / 256, 256, 0, stream>>>(dst, deg, E);
    deg_inv_k<<<(N + 255) / 256, 256, 0, stream>>>(deg, N);

    // ---- block 1 ----
    hipMemsetAsync(aggB, 0, featBytes, stream);
    agg_edges_k<<<edgeLaneBlocks, 256, 0, stream>>>(x, src, dst, aggB, E);
    sage_gemm_k<<<dim3(rowBlocks, 2), 128, 0, stream>>>(aggB, deg, x, WlT1, bl1, WrT1, hB, N, 128);
    ln_gelu_res_k<<<rowWaves8, 256, 0, stream>>>(hB, x, g1, b1, x1, N);

    // ---- block 2 ----
    hipMemsetAsync(aggB, 0, featBytes, stream);
    agg_edges_k<<<edgeLaneBlocks, 256, 0, stream>>>(x1, src, dst, aggB, E);
    sage_gemm_k<<<dim3(rowBlocks, 2), 128, 0, stream>>>(aggB, deg, x1, WlT2, bl2, WrT2, hB, N, 128);
    ln_gelu_res_k<<<rowWaves8, 256, 0, stream>>>(hB, x1, g2, b2, aggB, N);  // x2 -> aggB

    // ---- conv_out (Dout = 64) ----
    hipMemsetAsync(hB, 0, featBytes, stream);
    agg_edges_k<<<edgeLaneBlocks, 256, 0, stream>>>(aggB, src, dst, hB, E); // agg3 -> hB
    sage_gemm_k<<<dim3(rowBlocks, 1), 128, 0, stream>>>(hB, deg, aggB, WlT3, bl3, WrT3, x1, N, 64);
    gelu_ip_k<<<(int)(((long long)N * 64 + 255) / 256), 256, 0, stream>>>(x1, (long long)N * 64);

    // ---- classifier ----
    classifier_k<<<rowWaves8, 256, 0, stream>>>(x1, gc, bc, Wc, bcls, (float*)d_out, N);
}